// GLRFast_13606456394248
// MI455X (gfx1250) — compile-verified
//
#include <hip/hip_runtime.h>

// GLR 4-neighbor cross stencil: out = patchs - sum_e nbr_e * edge_w_e
// patchs:       [B, G, C, H, W] f32  (B=4, G=8, C=32, H=W=128)
// edge_weights: [B, G, 4, H, W] f32  (e order: up, left, right, down)
// node_degree:  unused in forward
//
// Memory-bound (≈1 FLOP/byte, ~143 MB -> ~6.1 us floor @ 23.3 TB/s).
// Strategy: per-(b,g,c) plane staged once into LDS via gfx1250 async
// global->LDS b128 DMA (ASYNCcnt), all 5 stencil taps served from LDS,
// 128-bit loads/stores everywhere.

#define G_DIM 8
#define C_DIM 32
#define H_DIM 128
#define W_DIM 128
#define HW_DIM (H_DIM * W_DIM)     // 16384 floats = 64 KB plane
#define W4_DIM (W_DIM / 4)         // 32 float4 per row
#define NTHREADS 256
#define STAGE_ITERS (HW_DIM / 4 / NTHREADS)  // 16 float4 per thread

__global__ __launch_bounds__(NTHREADS)
void glr_stencil_kernel(const float* __restrict__ patchs,
                        const float* __restrict__ edgew,
                        float* __restrict__ out)
{
    __shared__ float s_x[HW_DIM];   // one full 128x128 channel plane

    const int tid   = threadIdx.x;
    const int plane = blockIdx.x;          // = b*G*C + g*C + c
    const int bg    = plane / C_DIM;       // = b*G + g

    const float* gp = patchs + (size_t)plane * HW_DIM;
    const float* ge = edgew  + (size_t)bg * (4 * HW_DIM);
    float*       go = out    + (size_t)plane * HW_DIM;

    // ---- Phase 1: async DMA the whole plane into LDS (B128 per lane) ----
    const unsigned lds_base = (unsigned)(unsigned long long)(&s_x[0]);
#pragma unroll
    for (int k = 0; k < STAGE_ITERS; ++k) {
        const int q = tid + k * NTHREADS;                       // float4 index
        unsigned long long ga = (unsigned long long)(gp + q * 4);
        unsigned la = lds_base + (unsigned)(q * 16);
        asm volatile("global_load_async_to_lds_b128 %0, %1, off"
                     :: "v"(la), "v"(ga) : "memory");
    }

    // Overlap: prefetch this thread's first weight tiles into cache while
    // the async DMA is in flight (gfx1250 global_prefetch_b8).
    {
        const int q = tid;
        const int h = q >> 5;
        const int w = (q & (W4_DIM - 1)) << 2;
        const float* wrow = ge + h * W_DIM + w;
        __builtin_prefetch(wrow,              0, 1);
        __builtin_prefetch(wrow + 1 * HW_DIM, 0, 1);
        __builtin_prefetch(wrow + 2 * HW_DIM, 0, 1);
        __builtin_prefetch(wrow + 3 * HW_DIM, 0, 1);
    }

    asm volatile("s_wait_asynccnt 0" ::: "memory");
    __syncthreads();

    // ---- Phase 2: compute, all patchs taps from LDS ----
    const float4* s4 = reinterpret_cast<const float4*>(s_x);

#pragma unroll 4
    for (int k = 0; k < STAGE_ITERS; ++k) {
        const int q  = tid + k * NTHREADS;     // output float4 index in plane
        const int h  = q >> 5;                 // q / W4_DIM
        const int w4 = q & (W4_DIM - 1);
        const int w  = w4 << 2;

        const int hu = (h == 0)         ? 0 : h - 1;   // replicate pad
        const int hd = (h == H_DIM - 1) ? h : h + 1;

        const float4 c  = s4[h  * W4_DIM + w4];
        const float4 up = s4[hu * W4_DIM + w4];
        const float4 dn = s4[hd * W4_DIM + w4];

        const float le = (w4 == 0)           ? c.x : s_x[h * W_DIM + w - 1];
        const float re = (w4 == W4_DIM - 1)  ? c.w : s_x[h * W_DIM + w + 4];

        const float4 lf = make_float4(le,  c.x, c.y, c.z);   // x[w-1..w+2]
        const float4 rt = make_float4(c.y, c.z, c.w, re);    // x[w+1..w+4]

        const float* wbase = ge + h * W_DIM + w;
        const float4 w0 = *reinterpret_cast<const float4*>(wbase);              // up
        const float4 w1 = *reinterpret_cast<const float4*>(wbase + 1 * HW_DIM); // left
        const float4 w2 = *reinterpret_cast<const float4*>(wbase + 2 * HW_DIM); // right
        const float4 w3 = *reinterpret_cast<const float4*>(wbase + 3 * HW_DIM); // down

        float4 r;
        r.x = c.x - (up.x * w0.x + lf.x * w1.x + rt.x * w2.x + dn.x * w3.x);
        r.y = c.y - (up.y * w0.y + lf.y * w1.y + rt.y * w2.y + dn.y * w3.y);
        r.z = c.z - (up.z * w0.z + lf.z * w1.z + rt.z * w2.z + dn.z * w3.z);
        r.w = c.w - (up.w * w0.w + lf.w * w1.w + rt.w * w2.w + dn.w * w3.w);

        *reinterpret_cast<float4*>(go + h * W_DIM + w) = r;
    }
}

extern "C" void kernel_launch(void* const* d_in, const int* in_sizes, int n_in,
                              void* d_out, int out_size, void* d_ws, size_t ws_size,
                              hipStream_t stream)
{
    const float* patchs = (const float*)d_in[0];
    const float* edgew  = (const float*)d_in[1];
    // d_in[2] = node_degree, unused in forward
    float* out = (float*)d_out;

    const int planes = in_sizes[0] / HW_DIM;   // B*G*C (= 1024 for the ref shapes)

    glr_stencil_kernel<<<dim3(planes), dim3(NTHREADS), 0, stream>>>(patchs, edgew, out);
}